// StructuredDecoding_81836306858628
// MI455X (gfx1250) — compile-verified
//
#include <hip/hip_runtime.h>

// ---------------------------------------------------------------------------
// Soft-Viterbi (HMM forward + soft-backpointer backward) for MI455X (gfx1250).
//
// B=32, T=512, N=128.  Each step is recast as probability-space GEMMs:
//   forward:  v = exp(alpha_t - m) @ trans            (16x128 @ 128x128)
//             alpha_{t+1} = em + log(v) + m
//   backward: Z = a @ trans ; p_t = a * ((p_{t+1}/Z) @ trans^T)
// computed with V_WMMA_F32_16X16X4_F32 (exact f32 math).
//
// 2 workgroups x 256 threads (8 waves). Each WG owns 16 batches (M=16 = one
// WMMA tile row), each wave owns one 16-column N-tile, K=128 chained over 32
// WMMAs.  The per-wave B-operand tiles of the (constant) transition matrix
// are hoisted into registers (32 x v2f each), so the per-timestep critical
// path is only: LDS reads of the A operand + 32 chained WMMAs + epilogue.
// alpha is staged in d_out during the forward pass and overwritten in place
// with soft_path during the backward pass (no workspace needed).
// ---------------------------------------------------------------------------

typedef __attribute__((ext_vector_type(2))) float v2f;
typedef __attribute__((ext_vector_type(8))) float v8f;

#define NSTATE 128
#define TLEN   512
#define SROW   132          // padded LDS row stride (floats) -> conflict-free
#define BPG    16           // batches per workgroup (one WMMA M-tile)
#define NTHREADS 256
#define HMM_EPS 1e-10f

// LDS float offsets
#define OFF_TR    0                       // 128*132 = 16896
#define OFF_U     16896                   // 16*132  = 2112  (a = exp(alpha-m))
#define OFF_AL    19008                   // 16*132          (alpha_t)
#define OFF_P     21120                   // 16*132          (p_{t+1})
#define OFF_W     23232                   // 16*132          (p/Z, also log-prior staging)
#define OFF_AUX   25344                   // 320 scratch
#define LDS_FLOATS (25344 + 320)

__global__ __launch_bounds__(NTHREADS)
void hmm_soft_path_kernel(const float* __restrict__ trans_g,   // (N,N)
                          const float* __restrict__ logem,     // (B,T,N)
                          const float* __restrict__ prior_g,   // (N)
                          float* __restrict__ out)             // (B,T,N)
{
    extern __shared__ float lds[];
    float* tr  = lds + OFF_TR;
    float* uu  = lds + OFF_U;
    float* al  = lds + OFF_AL;
    float* pp  = lds + OFF_P;
    float* ww  = lds + OFF_W;
    float* aux = lds + OFF_AUX;

    const int tid   = threadIdx.x;
    const int lane  = tid & 31;
    const int wave  = tid >> 5;
    const int bbase = blockIdx.x * BPG;

    const int mrow = lane & 15;                 // WMMA M row / N col within tile
    const int koff = (lane < 16) ? 0 : 2;       // f32 16x16x4 A/B half-wave K offset
    const int n0   = wave * 16;                 // this wave's N-tile base

    // ---------------- transition: clamp + row-normalize (over 'next') -------
    for (int i = tid; i < NSTATE * NSTATE; i += NTHREADS) {
        int r = i >> 7, c = i & 127;
        tr[r * SROW + c] = fmaxf(trans_g[i], HMM_EPS);
    }
    __syncthreads();
    if (tid < NSTATE) {
        float s = 0.f;
        for (int c = 0; c < NSTATE; ++c) s += tr[tid * SROW + c];
        aux[tid] = s;
    }
    __syncthreads();
    for (int i = tid; i < NSTATE * NSTATE; i += NTHREADS) {
        int r = i >> 7, c = i & 127;
        tr[r * SROW + c] = tr[r * SROW + c] / aux[r];
    }
    __syncthreads();

    // ------ hoist this wave's B tile of trans into registers (forward) ------
    // btF[kk] = { trans[kk*4+koff][n0+mrow], trans[kk*4+koff+1][n0+mrow] }
    v2f btF[32];
#pragma unroll
    for (int kk = 0; kk < 32; ++kk) {
        int k = kk * 4 + koff;
        btF[kk].x = tr[k * SROW + n0 + mrow];
        btF[kk].y = tr[(k + 1) * SROW + n0 + mrow];
    }

    // ---------------- prior -> log prior (staged in ww[0..127]) -------------
    if (tid == 0) {
        float s = 0.f;
        for (int c = 0; c < NSTATE; ++c) s += fmaxf(prior_g[c], HMM_EPS);
        aux[256] = s;
    }
    __syncthreads();
    if (tid < NSTATE)
        ww[tid] = __logf(fmaxf(prior_g[tid], HMM_EPS) / aux[256]);
    __syncthreads();

    // ---------------- alpha[0] = log_prior + em[:,0] ------------------------
    for (int i = tid; i < BPG * NSTATE; i += NTHREADS) {
        int b = i >> 7, n = i & 127;
        float v = ww[n] + logem[((size_t)(bbase + b) * TLEN + 0) * NSTATE + n];
        al[b * SROW + n] = v;
        out[((size_t)(bbase + b) * TLEN + 0) * NSTATE + n] = v;
    }
    __syncthreads();

    // =================== forward scan: t = 1..T-1 ===========================
    for (int t = 1; t < TLEN; ++t) {
        // per-batch max of alpha (16 batches x 16 partials x 8 elems)
        {
            int b = tid >> 4, sub = tid & 15;
            float m = -3.4e38f;
            for (int j = 0; j < 8; ++j) m = fmaxf(m, al[b * SROW + sub * 8 + j]);
            aux[tid] = m;
        }
        __syncthreads();
        if (tid < BPG) {
            float m = aux[tid * 16];
            for (int s = 1; s < 16; ++s) m = fmaxf(m, aux[tid * 16 + s]);
            aux[256 + tid] = m;
        }
        __syncthreads();
        for (int i = tid; i < BPG * NSTATE; i += NTHREADS) {
            int b = i >> 7, j = i & 127;
            uu[b * SROW + j] = __expf(al[b * SROW + j] - aux[256 + b]);
        }
        __syncthreads();

        // v = u @ trans : each wave one 16x16 tile, K=128 via 32 chained WMMAs
        v8f acc = {};
#pragma unroll
        for (int kk = 0; kk < 32; ++kk) {
            v2f a;
            a.x = uu[mrow * SROW + kk * 4 + koff];
            a.y = uu[mrow * SROW + kk * 4 + koff + 1];
            acc = __builtin_amdgcn_wmma_f32_16x16x4_f32(
                false, a, false, btF[kk], (short)0, acc, false, false);
        }
        // epilogue: alpha_t = em + log(v) + m
        {
            int n = n0 + mrow;
            int rbase = (lane < 16) ? 0 : 8;
#pragma unroll
            for (int r = 0; r < 8; ++r) {
                int b = rbase + r;
                float em = logem[((size_t)(bbase + b) * TLEN + t) * NSTATE + n];
                float na = __logf(acc[r]) + aux[256 + b] + em;
                al[b * SROW + n] = na;
                out[((size_t)(bbase + b) * TLEN + t) * NSTATE + n] = na;
            }
        }
        __syncthreads();
    }

    // ------ hoist transposed B tile (backward only): btT[kk] = trans^T ------
    // btT[kk] = { trans[n0+mrow][kk*4+koff], trans[n0+mrow][kk*4+koff+1] }
    v2f btT[32];
#pragma unroll
    for (int kk = 0; kk < 32; ++kk) {
        int k = kk * 4 + koff;
        btT[kk].x = tr[(n0 + mrow) * SROW + k];
        btT[kk].y = tr[(n0 + mrow) * SROW + k + 1];
    }

    // =================== p[T-1] = softmax(alpha[T-1]) =======================
    {
        int b = tid >> 4, sub = tid & 15;
        float m = -3.4e38f;
        for (int j = 0; j < 8; ++j) m = fmaxf(m, al[b * SROW + sub * 8 + j]);
        aux[tid] = m;
        __syncthreads();
        if (tid < BPG) {
            float m2 = aux[tid * 16];
            for (int s = 1; s < 16; ++s) m2 = fmaxf(m2, aux[tid * 16 + s]);
            aux[256 + tid] = m2;
        }
        __syncthreads();
        float s = 0.f;
        for (int j = 0; j < 8; ++j) {
            float e = __expf(al[b * SROW + sub * 8 + j] - aux[256 + b]);
            pp[b * SROW + sub * 8 + j] = e;
            s += e;
        }
        aux[tid] = s;
        __syncthreads();
        if (tid < BPG) {
            float z = 0.f;
            for (int si = 0; si < 16; ++si) z += aux[tid * 16 + si];
            aux[256 + tid] = 1.f / z;
        }
        __syncthreads();
        for (int i = tid; i < BPG * NSTATE; i += NTHREADS) {
            int b2 = i >> 7, j = i & 127;
            float v = pp[b2 * SROW + j] * aux[256 + b2];
            pp[b2 * SROW + j] = v;
            out[((size_t)(bbase + b2) * TLEN + (TLEN - 1)) * NSTATE + j] = v;
        }
        __syncthreads();
    }

    // =================== backward scan: t = T-2..0 ==========================
    for (int t = TLEN - 2; t >= 0; --t) {
        // reload alpha[t] (stored in out during forward), then overwrite below
        for (int i = tid; i < BPG * NSTATE; i += NTHREADS) {
            int b = i >> 7, j = i & 127;
            al[b * SROW + j] = out[((size_t)(bbase + b) * TLEN + t) * NSTATE + j];
        }
        __syncthreads();
        {
            int b = tid >> 4, sub = tid & 15;
            float m = -3.4e38f;
            for (int j = 0; j < 8; ++j) m = fmaxf(m, al[b * SROW + sub * 8 + j]);
            aux[tid] = m;
        }
        __syncthreads();
        if (tid < BPG) {
            float m = aux[tid * 16];
            for (int s = 1; s < 16; ++s) m = fmaxf(m, aux[tid * 16 + s]);
            aux[256 + tid] = m;
        }
        __syncthreads();
        for (int i = tid; i < BPG * NSTATE; i += NTHREADS) {
            int b = i >> 7, j = i & 127;
            uu[b * SROW + j] = __expf(al[b * SROW + j] - aux[256 + b]);
        }
        __syncthreads();

        // Z = a @ trans ; w = p_{t+1} / Z
        v8f acc = {};
#pragma unroll
        for (int kk = 0; kk < 32; ++kk) {
            v2f a;
            a.x = uu[mrow * SROW + kk * 4 + koff];
            a.y = uu[mrow * SROW + kk * 4 + koff + 1];
            acc = __builtin_amdgcn_wmma_f32_16x16x4_f32(
                false, a, false, btF[kk], (short)0, acc, false, false);
        }
        {
            int n = n0 + mrow;
            int rbase = (lane < 16) ? 0 : 8;
#pragma unroll
            for (int r = 0; r < 8; ++r) {
                int b = rbase + r;
                ww[b * SROW + n] = pp[b * SROW + n] / acc[r];
            }
        }
        __syncthreads();

        // q = w @ trans^T ; p_t = a * q
        v8f acc2 = {};
#pragma unroll
        for (int kk = 0; kk < 32; ++kk) {
            v2f a;
            a.x = ww[mrow * SROW + kk * 4 + koff];
            a.y = ww[mrow * SROW + kk * 4 + koff + 1];
            acc2 = __builtin_amdgcn_wmma_f32_16x16x4_f32(
                false, a, false, btT[kk], (short)0, acc2, false, false);
        }
        {
            int j = n0 + mrow;
            int rbase = (lane < 16) ? 0 : 8;
#pragma unroll
            for (int r = 0; r < 8; ++r) {
                int b = rbase + r;
                float v = uu[b * SROW + j] * acc2[r];
                pp[b * SROW + j] = v;
                out[((size_t)(bbase + b) * TLEN + t) * NSTATE + j] = v;
            }
        }
        __syncthreads();
    }
}

extern "C" void kernel_launch(void* const* d_in, const int* in_sizes, int n_in,
                              void* d_out, int out_size, void* d_ws, size_t ws_size,
                              hipStream_t stream) {
    (void)n_in; (void)d_ws; (void)ws_size; (void)out_size;
    const float* trans  = (const float*)d_in[0];   // (N,N)
    const float* logem  = (const float*)d_in[1];   // (B,T,N)
    const float* prior  = (const float*)d_in[2];   // (N)
    float* out = (float*)d_out;                    // (B,T,N)

    const int B = in_sizes[1] / (TLEN * NSTATE);   // 32
    const int grid = B / BPG;                      // 2 workgroups (2 WGPs)
    const size_t shmem = (size_t)LDS_FLOATS * sizeof(float);  // ~100 KB < 320 KB/WGP

    hipLaunchKernelGGL(hmm_soft_path_kernel, dim3(grid), dim3(NTHREADS), shmem, stream,
                       trans, logem, prior, out);
}